// WordNgramsTokenizerLayer_77541339562499
// MI455X (gfx1250) — compile-verified
//
#include <hip/hip_runtime.h>
#include <stdint.h>

// ---- problem constants (match reference) ----
#define SWORDS      64          // words per row (S)
#define WORD_INTS   16          // bytes per word, stored 1 int32 per byte (L)
#define FNV_INIT    2166136261u
#define FNV_PRIME   16777619u
#define NGRAM_PRIME 116049371ull
#define BUCKET      2000000ull
#define VOCAB       200000ull

// ---- tiling ----
#define TILE_ROWS   4                       // batch rows per tile
#define TILE_WORDS  (TILE_ROWS * SWORDS)    // 256 words per tile = 1 per thread
#define PAD_INTS    20                      // 80B LDS stride/word (16B-aligned, anti-bank-conflict)
#define TILE_BYTES  (TILE_WORDS * WORD_INTS * 4)   // 16384 B of word_bytes per tile
#define NCHUNK_PT   4                       // 16B async chunks per thread per tile

__global__ __launch_bounds__(256, 2)
void wordngrams_kernel(const int* __restrict__ word_bytes,
                       const int* __restrict__ word_lens,
                       const int* __restrict__ word_counts,
                       float* __restrict__ out,
                       int B, int ntiles)
{
    // double-buffered staged bytes + per-tile word-hash exchange
    __shared__ uint32_t s_words[2][TILE_WORDS * PAD_INTS];   // 2 x 20 KB
    __shared__ uint32_t s_hash[TILE_WORDS];                  // 1 KB

    const int tid = (int)threadIdx.x;
    const uint64_t wb_base     = (uint64_t)(uintptr_t)word_bytes;
    const uint64_t total_bytes = (uint64_t)B * (uint64_t)(SWORDS * WORD_INTS * 4);

    uint32_t ldsA[2];
    ldsA[0] = (uint32_t)(uintptr_t)(&s_words[0][0]);   // low 32 bits of generic ptr == LDS offset
    ldsA[1] = (uint32_t)(uintptr_t)(&s_words[1][0]);

    // Issue one tile's worth of async global->LDS copies: 1024 chunks of 16B,
    // lane-contiguous in global (perfectly coalesced), word-padded in LDS.
    auto issue_tile = [&](int tile, int buf) {
        const uint64_t tbase = (uint64_t)tile * (uint64_t)TILE_BYTES;
        const uint64_t gbase = wb_base + tbase;     // uniform -> SGPR pair
        const uint32_t lbase = ldsA[buf];
#pragma unroll
        for (int i = 0; i < NCHUNK_PT; ++i) {
            const int c = tid + i * 256;            // chunk id 0..1023
            uint32_t goff = (uint32_t)(c * 16);
            if (tbase + (uint64_t)goff + 16ull > total_bytes) goff = 0;  // tail clamp (safe, unused rows)
            const uint32_t loff = lbase + (uint32_t)((c >> 2) * (PAD_INTS * 4) + (c & 3) * 16);
            asm volatile("global_load_async_to_lds_b128 %0, %1, %2"
                         :: "v"(loff), "v"(goff), "s"(gbase) : "memory");
        }
    };

    int tile = (int)blockIdx.x;
    if (tile >= ntiles) return;

    issue_tile(tile, 0);                            // prologue prefetch
    int it = 0;
    for (; tile < ntiles; tile += (int)gridDim.x, ++it) {
        const int  next     = tile + (int)gridDim.x;
        const bool has_next = next < ntiles;
        if (has_next) {
            issue_tile(next, (it + 1) & 1);         // keep next tile's DMA in flight
            // 8 async loads outstanding per wave; in-order completion =>
            // cnt<=4 retires exactly the current tile's 4 loads.
            asm volatile("s_wait_asynccnt 4" ::: "memory");
        } else {
            asm volatile("s_wait_asynccnt 0" ::: "memory");
        }
        __syncthreads();                            // all waves' current-tile DMA visible

        const uint32_t* wbuf  = &s_words[it & 1][0];
        const int       gword = tile * TILE_WORDS + tid;   // global word index = row*64 + s
        const int       row   = gword >> 6;

        // ---- stage 1: FNV-1a over this thread's word (from LDS) ----
        uint32_t h = FNV_INIT;
        if (row < B) {
            const int len = word_lens[gword];              // 1..16
            const uint4* p4 = (const uint4*)(wbuf + tid * PAD_INTS);  // 80B-aligned
            const uint4 q0 = p4[0], q1 = p4[1], q2 = p4[2], q3 = p4[3];
            uint32_t v[16] = { q0.x, q0.y, q0.z, q0.w,
                               q1.x, q1.y, q1.z, q1.w,
                               q2.x, q2.y, q2.z, q2.w,
                               q3.x, q3.y, q3.z, q3.w };
#pragma unroll
            for (int i = 0; i < 16; ++i) {
                const uint32_t nh = (h ^ v[i]) * FNV_PRIME;
                h = (i < len) ? nh : h;
            }
        }
        s_hash[tid] = h;
        __syncthreads();                            // hashes visible to neighbors

        // ---- stage 2: rolling n-gram hash over word positions ----
        if (row < B) {
            const int s   = tid & (SWORDS - 1);
            const int rb  = tid & ~(SWORDS - 1);    // tile-local row base
            const int cnt = word_counts[row];
            // sign-extend int32 -> uint64 (matches reference cast chain)
            uint64_t hh = (uint64_t)(int64_t)(int32_t)h;
            const long long ob = (long long)row * (4 * SWORDS) + s;
#pragma unroll
            for (int n = 2; n <= 5; ++n) {
                int j = s + n - 1;                  // contributing position
                if (j > SWORDS - 1) j = SWORDS - 1; // clamp; such slots are masked anyway
                const uint64_t xs = (uint64_t)(int64_t)(int32_t)s_hash[rb + j];
                hh = hh * NGRAM_PRIME + xs;         // mod 2^64
                const uint64_t idx = hh % BUCKET + VOCAB;      // < 2.2e6 < 2^24: exact in fp32
                const bool valid = (s + n) <= cnt;
                out[ob + (long long)(n - 2) * SWORDS] = valid ? (float)(uint32_t)idx : 0.0f;
            }
        }
        __syncthreads();   // protect s_hash + byte buffer before next iteration's writers
    }
}

extern "C" void kernel_launch(void* const* d_in, const int* in_sizes, int n_in,
                              void* d_out, int out_size, void* d_ws, size_t ws_size,
                              hipStream_t stream) {
    const int* word_bytes  = (const int*)d_in[0];   // [B, 64, 16] int32 (one utf-8 byte per int)
    const int* word_lens   = (const int*)d_in[1];   // [B, 64] int32
    const int* word_counts = (const int*)d_in[2];   // [B] int32
    float* out = (float*)d_out;                     // [B, 4, 64]; all values < 2^24, exact in fp32

    const int B      = in_sizes[2];                 // word_counts has B elements
    const int ntiles = (B + TILE_ROWS - 1) / TILE_ROWS;
    int grid = ntiles < 1024 ? ntiles : 1024;       // grid-stride over tiles, double-buffered DMA
    if (grid < 1) grid = 1;

    wordngrams_kernel<<<grid, 256, 0, stream>>>(word_bytes, word_lens, word_counts,
                                                out, B, ntiles);
}